// ParallelNet_65850438582500
// MI455X (gfx1250) — compile-verified
//
#include <hip/hip_runtime.h>
#include <hip/hip_bf16.h>

// ---------------------------------------------------------------------------
// ParallelNet on MI455X (gfx1250, wave32, WMMA).
// out[512,10] = (concat_p (h @ Wp[p].T + bp[p])) @ W_out.T + b_out,
//   h = x @ W_in.T + b_in.
// Fused per-layer: T_p = h @ Wp[p].T  (FP32 WMMA 16x16x4, K fully unrolled,
// all 50 operand loads issued before the 25-WMMA accumulation chain),
// then out += T_p @ Wout_p.T  (VALU contraction, 10 outputs).
// Bias terms are batch-independent -> folded into base[o] once.
// Bandwidth-bound: Wp (200 MB) streams once; h (200 KB) stays in L2.
// ---------------------------------------------------------------------------

typedef __attribute__((ext_vector_type(2))) float v2f;
typedef __attribute__((ext_vector_type(8))) float v8f;

#define BATCH 512
#define DIM   100
#define NLAYER 5000
#define NOUT  10

#define ROWS  64          // batch rows per block
#define LPER  20          // layers per block
#define TPAD  112         // sT row stride (pads 100 -> 112)
#define KSTEPS 25         // 100 / 4

// ---------------- h = x @ W_in.T + b_in  -> ws.h [512][100] ----------------
__global__ __launch_bounds__(256) void h_kernel(
    const float* __restrict__ x, const float* __restrict__ W_in,
    const float* __restrict__ b_in, float* __restrict__ h) {
  int i = blockIdx.x * 256 + threadIdx.x;
  if (i >= BATCH * DIM) return;
  int b = i / DIM, d = i % DIM;
  const float* xr = x + b * DIM;
  const float* wr = W_in + d * DIM;
  float s = b_in[d];
#pragma unroll 4
  for (int k = 0; k < DIM; ++k) s = fmaf(xr[k], wr[k], s);
  h[i] = s;
}

// ------------- base[o] = b_out[o] + W_out[o,:] . bp_flat[:] ----------------
__global__ __launch_bounds__(256) void base_kernel(
    const float* __restrict__ W_out, const float* __restrict__ bp,
    const float* __restrict__ b_out, float* __restrict__ base) {
  __shared__ float red[256];
  int o = blockIdx.x;                       // 0..9
  const float* wr = W_out + (size_t)o * (NLAYER * DIM);
  float s = 0.f;
  for (int k = threadIdx.x; k < NLAYER * DIM; k += 256)
    s = fmaf(wr[k], bp[k], s);
  red[threadIdx.x] = s;
  __syncthreads();
  for (int off = 128; off > 0; off >>= 1) {
    if (threadIdx.x < off) red[threadIdx.x] += red[threadIdx.x + off];
    __syncthreads();
  }
  if (threadIdx.x == 0) base[o] = red[0] + b_out[o];
}

// ---------------------- out[b][o] = base[o] --------------------------------
__global__ __launch_bounds__(256) void init_out_kernel(
    const float* __restrict__ base, float* __restrict__ out) {
  int i = blockIdx.x * 256 + threadIdx.x;
  if (i < BATCH * NOUT) out[i] = base[i % NOUT];
}

// ---- one 16x16 T-tile: C = A(16x100) x B(100x16), FP32 WMMA 16x16x4 -------
// hA/bB already include this lane's +half*2 K offset; per k-step each lane
// supplies a K-pair (A layout: lanes 0-15 K=0..1, lanes 16-31 K=2..3; B
// mirrored over columns).  MASKED path zeroes B cols >= DIM (tail N-tile).
template <bool MASKED>
__device__ __forceinline__ v8f tile_gemm(const float* __restrict__ hA,
                                         const float* __restrict__ bB,
                                         float fmask) {
  v2f a[KSTEPS], b[KSTEPS];
#pragma unroll
  for (int k = 0; k < KSTEPS; ++k) {        // 50 independent global_load_b64
    a[k] = *(const v2f*)(hA + 4 * k);
    v2f bv = *(const v2f*)(bB + 4 * k);
    b[k] = MASKED ? (bv * fmask) : bv;
  }
  v8f c = {};
#pragma unroll
  for (int k = 0; k < KSTEPS; ++k)          // dense 25-WMMA accumulation chain
    c = __builtin_amdgcn_wmma_f32_16x16x4_f32(
        false, a[k], false, b[k], (short)0, c, false, false);
  return c;
}

// --------------------------- main fused kernel ------------------------------
// grid = (NLAYER/LPER) * (BATCH/ROWS) blocks, 256 threads (8 waves).
// block -> (layer group, 64-row batch chunk).
__global__ __launch_bounds__(256) void parallel_layers_kernel(
    const float* __restrict__ h, const float* __restrict__ Wp,
    const float* __restrict__ W_out, float* __restrict__ out) {
  __shared__ float sT[ROWS * TPAD];     // per-layer T tile buffer  (28.7 KB)
  __shared__ float swo[NOUT * DIM];     // W_out slice for layer p  ( 4.0 KB)
  __shared__ float sacc[ROWS * NOUT];   // running output accum     ( 2.5 KB)

  const int tid  = threadIdx.x;
  const int wave = tid >> 5;
  const int lane = tid & 31;
  const int half = lane >> 4;           // 0: K=0..1 / M=0..7 ; 1: K=2..3 / M=8..15
  const int l15  = lane & 15;

  const int chunk = blockIdx.x & 7;     // 8 chunks of 64 rows
  const int group = blockIdx.x >> 3;    // 250 groups of 20 layers
  const int row0  = chunk * ROWS;

  for (int i = tid; i < ROWS * NOUT; i += 256) sacc[i] = 0.f;
  __syncthreads();

  for (int li = 0; li < LPER; ++li) {
    const int p = group * LPER + li;
    const float* wp = Wp + (size_t)p * (DIM * DIM);

    // warm L2/WGP$ for this layer's weights (global_prefetch_b8)
    for (int i = tid * 32; i < DIM * DIM; i += 256 * 32)
      __builtin_prefetch((const void*)(wp + i), 0, 1);

    // stage this layer's W_out slice: swo[o*100 + j] = W_out[o][p*100 + j]
    for (int i = tid; i < NOUT * DIM; i += 256) {
      int o = i / DIM, j = i % DIM;
      swo[i] = W_out[(size_t)o * (NLAYER * DIM) + (size_t)p * DIM + j];
    }

    // ---- stage 2: T = h_chunk @ Wp[p].T via FP32 WMMA 16x16x4 ----
    // 4 M-tiles x 7 N-tiles = 28 tile jobs over 8 waves
    for (int job = wave; job < 28; job += 8) {
      const int mt = job / 7;
      const int nt = job % 7;           // wave-uniform -> uniform branch below
      const int mrow = row0 + mt * 16;
      const int colBase = nt * 16;

      const int arow = mrow + l15;                  // A: row per lane
      const int brow = colBase + l15;               // B: output col j -> Wp row
      const float fmask = (brow < DIM) ? 1.0f : 0.0f;
      const float* hA = h  + (size_t)arow * DIM + half * 2;
      const float* bB = wp + (size_t)(brow < DIM ? brow : 0) * DIM + half * 2;

      v8f c;
      if (nt == 6) c = tile_gemm<true >(hA, bB, fmask);   // tail cols masked
      else         c = tile_gemm<false>(hA, bB, 1.0f);    // hot path, no mask

      // C/D layout: lanes 0-15 -> M = r, lanes 16-31 -> M = 8 + r; N = l15
      const int lrow = mt * 16 + half * 8;
      const int j = colBase + l15;            // < TPAD always
#pragma unroll
      for (int r = 0; r < 8; ++r) sT[(lrow + r) * TPAD + j] = c[r];
    }
    __syncthreads();

    // ---- stage 3: sacc[m][o] += T[m][:] . swo[o][:]  (one owner per (m,o)) ----
    for (int t = tid; t < ROWS * NOUT; t += 256) {
      const int m = t / NOUT, o = t % NOUT;
      const float* tr = &sT[m * TPAD];
      const float* wr = &swo[o * DIM];
      float s = 0.f;
#pragma unroll 4
      for (int k = 0; k < DIM; ++k) s = fmaf(tr[k], wr[k], s);
      sacc[t] += s;
    }
    __syncthreads();   // protect sT/swo before next layer overwrites them
  }

  // flush: one atomic per (row, o) per block (global_atomic_add_f32)
  for (int t = tid; t < ROWS * NOUT; t += 256) {
    const int m = t / NOUT, o = t % NOUT;
    atomicAdd(&out[(size_t)(row0 + m) * NOUT + o], sacc[t]);
  }
}

// ---------------------------------------------------------------------------
extern "C" void kernel_launch(void* const* d_in, const int* in_sizes, int n_in,
                              void* d_out, int out_size, void* d_ws, size_t ws_size,
                              hipStream_t stream) {
  const float* x     = (const float*)d_in[0];   // [512,100]
  const float* W_in  = (const float*)d_in[1];   // [100,100]
  const float* b_in  = (const float*)d_in[2];   // [100]
  const float* Wp    = (const float*)d_in[3];   // [5000,100,100]
  const float* bp    = (const float*)d_in[4];   // [5000,100]
  const float* W_out = (const float*)d_in[5];   // [10,500000]
  const float* b_out = (const float*)d_in[6];   // [10]
  float* out = (float*)d_out;                   // [512,10]

  float* ws_h    = (float*)d_ws;                // 51200 floats
  float* ws_base = ws_h + BATCH * DIM;          // 10 floats

  h_kernel<<<(BATCH * DIM + 255) / 256, 256, 0, stream>>>(x, W_in, b_in, ws_h);
  base_kernel<<<NOUT, 256, 0, stream>>>(W_out, bp, b_out, ws_base);
  init_out_kernel<<<(BATCH * NOUT + 255) / 256, 256, 0, stream>>>(ws_base, out);

  const int grid = (NLAYER / LPER) * (BATCH / ROWS);  // 250 * 8 = 2000
  parallel_layers_kernel<<<grid, 256, 0, stream>>>(ws_h, Wp, W_out, out);
}